// MonarchLayer_9165460210358
// MI455X (gfx1250) — compile-verified
//
#include <hip/hip_runtime.h>

// Monarch layer, fused, fp32 WMMA (V_WMMA_F32_16X16X4_F32), CDNA5/gfx1250.

typedef __attribute__((ext_vector_type(2))) float v2f;
typedef __attribute__((ext_vector_type(4))) float f4;
typedef __attribute__((ext_vector_type(8))) float v8f;

#if defined(__HIP_DEVICE_COMPILE__) && !__has_builtin(__builtin_amdgcn_wmma_f32_16x16x4_f32)
#error "missing __builtin_amdgcn_wmma_f32_16x16x4_f32 on this toolchain (device pass)"
#endif

#define DIM   4096
#define MB    64          // monarch block size m
#define BT    16          // batch rows per workgroup
#define S_R   66          // h1 row stride (floats): 8*S_R mod 64 == 16 -> write halves disjoint
#define S_K   1072        // h1 k-block stride (floats): 2*S_K mod 64 == 32 -> read halves disjoint
#define XROWS 68          // x-tile row stride: bank = 4*row + K, conflict-free; rows 16B-aligned
#define XT_S  (16 * XROWS)
#define H1_FLOATS (64 * S_K)
#define SMEM_FLOATS (H1_FLOATS + 4 * XT_S)

// Wave-uniform element offset forced scalar (SGPR) without breaking the base
// pointer's global-addrspace provenance.
__device__ __forceinline__ unsigned uniform_off(int e) {
    return (unsigned)__builtin_amdgcn_readfirstlane(e);
}

__global__ __launch_bounds__(512, 1)
void monarch_f32_wmma(const float* __restrict__ x,
                      const float* __restrict__ Lw,
                      const float* __restrict__ Rw,
                      const float* __restrict__ bias,
                      float* __restrict__ y)
{
    extern __shared__ float smem[];
    float* h1  = smem;              // h1[k][row][j] : k*S_K + row*S_R + j
    float* xts = smem + H1_FLOATS;  // 4 groups of 16x64 X tiles (padded)

    const int  tid  = threadIdx.x;
    const int  w    = tid >> 5;     // wave 0..15
    const int  lane = tid & 31;
    const int  n    = lane & 15;    // N / M index within 16-wide tile
    const int  hi   = lane >> 4;    // upper-half lane -> K offset +2, M offset +8
    const long b0   = (long)blockIdx.x * BT;

    // ---------------- Stage 1: h1[b,k,j] = sum_i x[b,k,i] * R[k,i,j] -----
    const int g  = w >> 2;          // which k of the 4 processed per phase
    const int nt = w & 3;           // 16-wide j tile
    float* xt = xts + g * XT_S;
    // cooperative x-tile load mapping: 128 threads x 8 floats = 16x64 tile
    const int fidx = (((w & 3) << 5) | lane) << 3;
    const int xrow = fidx >> 6;
    const int xcol = fidx & 63;
    const float* xrowbase = x + (b0 + xrow) * DIM;   // global base + 32-bit lane offset

    const unsigned bcol = (unsigned)((nt << 4) + n); // B lane column within 64-wide row

    for (int p = 0; p < 16; ++p) {
        const int k = (p << 2) + g;
        {   // coalesced global -> LDS stage of X[b0..b0+15][k*64 .. +63]
            const float* src = xrowbase + k * MB + xcol;
            f4 q0 = __builtin_nontemporal_load((const f4*)(src));
            f4 q1 = __builtin_nontemporal_load((const f4*)(src + 4));
            f4* d = (f4*)(xt + xrow * XROWS + xcol); // XROWS%4==0, xcol%8==0 -> 16B aligned
            d[0] = q0;
            d[1] = q1;
        }
        __syncthreads();

        // One per-lane 64-bit base per phase; every B load below is this base
        // plus a compile-time immediate (K0*256(+256) bytes, < 24-bit IOFFSET).
        const float* Rb = Rw + uniform_off(k * (MB * MB)) + bcol;
        v8f acc = {};
#pragma unroll
        for (int kk = 0; kk < 16; ++kk) {
            const int K0 = (kk << 2) + (hi << 1);   // K = K0, K0+1 for this lane
            v2f a, b;
            a.x = xt[n * XROWS + K0];               // A[M=n][K]  (ds_load_2addr pair)
            a.y = xt[n * XROWS + K0 + 1];
            b.x = Rb[K0 * MB];                      // B[K][N]  base + imm
            b.y = Rb[(K0 + 1) * MB];
            acc = __builtin_amdgcn_wmma_f32_16x16x4_f32(
                      false, a, false, b, (short)0, acc, false, false);
        }
        {   // D[row=v+8*hi][j=nt*16+n] -> h1[k][row][j]  (conflict-free)
            float* d = h1 + k * S_K + (nt << 4) + n + (hi ? 8 * S_R : 0);
#pragma unroll
            for (int v = 0; v < 8; ++v) d[v * S_R] = acc[v];
        }
        __syncthreads();
    }

    // ---------------- Stage 2: y[b,k'*64+j] = sum_k h1[b,k,j] * L[j,k,k'] + bias
    for (int t = 0; t < 16; ++t) {
        const int j   = (w << 2) + (t & 3);   // 0..63  (wave-uniform)
        const int nt2 = t >> 2;               // 16-wide k' tile
        const unsigned bcol2 = (unsigned)((nt2 << 4) + n);
        const float* Lb = Lw + uniform_off(j * (MB * MB)) + bcol2;  // one base per task
        v8f acc = {};
#pragma unroll
        for (int kk = 0; kk < 16; ++kk) {
            const int K0 = (kk << 2) + (hi << 1);   // K = k index
            v2f a, b;
            a.x = h1[K0 * S_K + n * S_R + j];       // A[M=n][K] = h1[b,K,j] (transpose via layout)
            a.y = h1[(K0 + 1) * S_K + n * S_R + j];
            b.x = Lb[K0 * MB];                      // base + imm
            b.y = Lb[(K0 + 1) * MB];
            acc = __builtin_amdgcn_wmma_f32_16x16x4_f32(
                      false, a, false, b, (short)0, acc, false, false);
        }
        const int   col = ((nt2 << 4) + n) * MB + j;   // k'*64 + j
        const float bv  = bias[col];
        float* dst = y + (b0 + (hi ? 8 : 0)) * DIM + col;
#pragma unroll
        for (int v = 0; v < 8; ++v) dst[(long)v * DIM] = acc[v] + bv;  // merges in L2
    }
}

extern "C" void kernel_launch(void* const* d_in, const int* in_sizes, int n_in,
                              void* d_out, int out_size, void* d_ws, size_t ws_size,
                              hipStream_t stream) {
    const float* x    = (const float*)d_in[0];
    const float* L    = (const float*)d_in[1];
    const float* R    = (const float*)d_in[2];
    const float* bias = (const float*)d_in[3];
    float*       y    = (float*)d_out;

    const int batch = in_sizes[0] / DIM;          // 16384
    dim3 grid(batch / BT);                        // 1024 workgroups
    const size_t shmem = (size_t)SMEM_FLOATS * sizeof(float);  // ~285 KB of 320 KB LDS
    monarch_f32_wmma<<<grid, dim3(512), shmem, stream>>>(x, L, R, bias, y);
}